// MLoss_498216206539
// MI455X (gfx1250) — compile-verified
//
#include <hip/hip_runtime.h>

typedef __attribute__((ext_vector_type(4))) float v4f;
typedef __attribute__((ext_vector_type(2))) float v2f;
typedef __attribute__((ext_vector_type(8))) float v8f;

#define NBLK 1024
#define NTHR 256
#define NWAVES (NTHR / 32)

// gfx1250 has V_TANH_F32 (one TRANS op). sigmoid(x) = 0.5 + 0.5*tanh(x/2)
// costs 1 TRANS instead of exp+rcp (2 TRANS). Guarded so compilation never
// regresses if the builtin spelling differs in this toolchain.
#if __has_builtin(__builtin_amdgcn_tanhf)
  #define FAST_TANHF(v) __builtin_amdgcn_tanhf(v)
#elif __has_builtin(__builtin_amdgcn_tanh_f32)
  #define FAST_TANHF(v) __builtin_amdgcn_tanh_f32(v)
#endif

__device__ __forceinline__ float fast_sigmoid(float v) {
#ifdef FAST_TANHF
    return __builtin_fmaf(0.5f, FAST_TANHF(0.5f * v), 0.5f);
#else
    float e = __expf(-v);
    return __builtin_amdgcn_rcpf(1.0f + e);
#endif
}

__device__ __forceinline__ float cell_contrib(const float* xc, const float* yc) {
    float o0 = fast_sigmoid(xc[0]);
    float o1 = fast_sigmoid(xc[1]);
    float o2 = fast_sigmoid(xc[2]);
    float o3 = __expf(xc[3]) * (1.0f / 416.0f);
    float o4 = __expf(xc[4]) * (1.0f / 416.0f);
    float d0 = yc[0] - o0, d1 = yc[1] - o1, d2 = yc[2] - o2;
    float d3 = yc[3] - o3, d4 = yc[4] - o4;
    float sq = d0 * d0 + d1 * d1 + d2 * d2 + d3 * d3 + d4 * d4;
    // m*||y-out||^2 + 0.1*(1-m)*o0^2
    return (yc[0] > 0.5f) ? sq : 0.1f * o0 * o0;
}

__device__ __forceinline__ float readlane_f32(float v, int lane) {
    return __uint_as_float(__builtin_amdgcn_readlane(__float_as_uint(v), lane));
}

// Streaming masked-MSE partial reduction. Each thread handles groups of
// 4 cells (20 floats = 5 x b128, 16B aligned, TH=NT since the 218MB stream
// exceeds the 192MB L2 and is never reused). Per-wave sum via one
// V_WMMA_F32_16X16X4_F32 with B = ones.
__global__ void __launch_bounds__(NTHR)
mloss_partial(const float* __restrict__ x, const float* __restrict__ y,
              float* __restrict__ ws, long long ngroups, long long ncells) {
    float acc = 0.0f;
    long long tid    = (long long)blockIdx.x * NTHR + threadIdx.x;
    long long stride = (long long)gridDim.x * NTHR;

    for (long long g = tid; g < ngroups; g += stride) {
        const v4f* xp = (const v4f*)(x + g * 20);
        const v4f* yp = (const v4f*)(y + g * 20);
        float xf[20], yf[20];
#pragma unroll
        for (int i = 0; i < 5; ++i) {
            v4f xv = __builtin_nontemporal_load(xp + i);   // global_load_b128 NT
            v4f yv = __builtin_nontemporal_load(yp + i);
            xf[4 * i + 0] = xv.x; xf[4 * i + 1] = xv.y;
            xf[4 * i + 2] = xv.z; xf[4 * i + 3] = xv.w;
            yf[4 * i + 0] = yv.x; yf[4 * i + 1] = yv.y;
            yf[4 * i + 2] = yv.z; yf[4 * i + 3] = yv.w;
        }
#pragma unroll
        for (int c = 0; c < 4; ++c)
            acc += cell_contrib(xf + 5 * c, yf + 5 * c);
    }

    // Scalar tail (cells not divisible by 4) — global thread 0 only;
    // control flow reconverges before the WMMA (EXEC all-ones there).
    if (tid == 0) {
        for (long long c = ngroups * 4; c < ncells; ++c)
            acc += cell_contrib(x + c * 5, y + c * 5);
    }

    // ---- wave32 reduction with one V_WMMA_F32_16X16X4_F32 ----
    // A (16x4 f32) = exactly 2 VGPRs/lane: partial in slot 0, zero in slot 1.
    // B = all ones  ->  D[m][n] = sum_k A[m][k]; every A slot is some lane's acc.
    // All columns of D are equal: lanes 0..15 hold rows 0..7 in VGPRs 0..7,
    // lanes 16..31 hold rows 8..15. So sum(D VGPRs) at lane 0 + lane 16 = wave sum.
    v2f a; a.x = acc; a.y = 0.0f;
    v2f b; b.x = 1.0f; b.y = 1.0f;
    v8f cm = {};
    cm = __builtin_amdgcn_wmma_f32_16x16x4_f32(
        /*neg_a=*/false, a, /*neg_b=*/false, b,
        /*c_mod=*/(short)0, cm, /*reuse_a=*/false, /*reuse_b=*/false);
    float half = ((cm[0] + cm[1]) + (cm[2] + cm[3])) +
                 ((cm[4] + cm[5]) + (cm[6] + cm[7]));
    float wave_sum = readlane_f32(half, 0) + readlane_f32(half, 16);

    __shared__ float smem[NWAVES];
    int lane = threadIdx.x & 31;
    int wave = threadIdx.x >> 5;
    if (lane == 0) smem[wave] = wave_sum;
    __syncthreads();
    if (threadIdx.x == 0) {
        float bsum = 0.0f;
#pragma unroll
        for (int i = 0; i < NWAVES; ++i) bsum += smem[i];
        ws[blockIdx.x] = bsum;
    }
}

// Deterministic fixed-order final reduce of the NBLK per-block partials.
__global__ void __launch_bounds__(NTHR)
mloss_final(const float* __restrict__ ws, float* __restrict__ out, int n) {
    __shared__ float s[NTHR];
    float a = 0.0f;
    for (int i = threadIdx.x; i < n; i += NTHR) a += ws[i];
    s[threadIdx.x] = a;
    __syncthreads();
    for (int w = NTHR / 2; w > 0; w >>= 1) {
        if ((int)threadIdx.x < w) s[threadIdx.x] += s[threadIdx.x + w];
        __syncthreads();
    }
    if (threadIdx.x == 0) out[0] = s[0];
}

extern "C" void kernel_launch(void* const* d_in, const int* in_sizes, int n_in,
                              void* d_out, int out_size, void* d_ws, size_t ws_size,
                              hipStream_t stream) {
    const float* x = (const float*)d_in[0];   // [512, 10647, 5] f32
    const float* y = (const float*)d_in[1];   // [512, 10647, 5] f32
    float* out = (float*)d_out;               // scalar f32
    float* ws  = (float*)d_ws;                // NBLK partials (4 KB)

    long long nelem   = (long long)in_sizes[0];
    long long ncells  = nelem / 5;
    long long ngroups = ncells / 4;

    mloss_partial<<<NBLK, NTHR, 0, stream>>>(x, y, ws, ngroups, ncells);
    mloss_final<<<1, NTHR, 0, stream>>>(ws, out, NBLK);
}